// Attention_71176198029467
// MI455X (gfx1250) — compile-verified
//
#include <hip/hip_runtime.h>
#include <math.h>

// CDNA5 (gfx1250) wave32 fp32-WMMA attention:
//   x: [B, L, D=256, H=32, W=32] fp32  ->  xf: [B, L, N=1024, D=256] (logical)
//   out[b,l] = softmax(beta * xf xf^T) xf
// One workgroup (4 wave32) per (b*l, 16-row tile). Scores for the 16-row
// block (16 x 1024 f32) are materialized in LDS (64 KB, fits CDNA5's 320 KB
// WGP LDS), softmax is done exactly (max-subtracted) across full rows, then
// O = P*V. All matmuls use V_WMMA_F32_16X16X4_F32 for exact fp32 fidelity
// with the fp32 reference (f16 inputs would perturb softmax by ~1%).

typedef float v2f __attribute__((ext_vector_type(2)));
typedef float v8f __attribute__((ext_vector_type(8)));

#define N_TOK 1024
#define D_DIM 256
#define S_PITCH 1028   // 1024 + 4 pad -> row stride % 64 banks == 4 (conflict-free)
#define Q_PITCH 260    // 256 + 4 pad

#define S_OFF    0
#define Q_OFF    (16 * S_PITCH)
#define RED_OFF  (Q_OFF + 16 * Q_PITCH)
#define RMAX_OFF (RED_OFF + 16 * 8)
#define RINV_OFF (RMAX_OFF + 16)
#define SMEM_FLOATS (RINV_OFF + 16)

__global__ __launch_bounds__(128) void attn_f32wmma_kernel(
    const float* __restrict__ x, const int* __restrict__ beta,
    float* __restrict__ out) {
  extern __shared__ float smem[];
  float* Slds = smem + S_OFF;   // [16][S_PITCH] scores, then exp(scores)
  float* Qlds = smem + Q_OFF;   // [16][Q_PITCH] query tile
  float* red  = smem + RED_OFF; // [16][8] partial row reductions
  float* rmax = smem + RMAX_OFF;
  float* rinv = smem + RINV_OFF;

  const int tid  = threadIdx.x;
  const int wv   = tid >> 5;     // wave id 0..3 (wave32)
  const int lane = tid & 31;
  const int half = lane >> 4;    // 0: lanes 0-15, 1: lanes 16-31
  const int lm   = lane & 15;
  const int koff = half << 1;    // A/B 32-bit operand layout: K = {0,1} | {2,3}

  const int m0 = blockIdx.x * 16;               // query-row tile
  const long long bl = blockIdx.y;              // fused (b,l)
  const float* slice = x + bl * (long long)(D_DIM * N_TOK);  // [256][1024]: xf[n][d] = slice[d*1024+n]
  float* outp = out + bl * (long long)(N_TOK * D_DIM);       // [1024][256]

  const float betaF = (float)beta[0];

  // ---- stage Q tile (16 rows x 256 d) into LDS, coalesced over n ----
  for (int idx = tid; idx < 16 * D_DIM; idx += 128) {
    const int m = idx & 15, d = idx >> 4;
    Qlds[m * Q_PITCH + d] = slice[d * N_TOK + m0 + m];
  }
  __syncthreads();

  // ---- phase 1: S = Q K^T ; wave wv owns key columns [wv*256, wv*256+256) ----
  for (int j = 0; j < 16; ++j) {
    const int n0 = wv * 256 + j * 16;
    v8f acc = {};
    for (int k = 0; k < D_DIM; k += 4) {
      // A (16x4): lane lm = row m, this half's K pair from LDS (8B load)
      v2f a = *(const v2f*)&Qlds[lm * Q_PITCH + k + koff];
      // B (4x16): B[kk][n] = K[n0+n][kk] = slice[kk*1024 + n0+n]
      v2f b;
      b.x = slice[(k + koff) * N_TOK + n0 + lm];
      b.y = slice[(k + koff + 1) * N_TOK + n0 + lm];
      acc = __builtin_amdgcn_wmma_f32_16x16x4_f32(false, a, false, b,
                                                  (short)0, acc, false, false);
    }
    // C/D layout: VGPR r -> row r (lanes 0-15) / row r+8 (lanes 16-31)
    for (int r = 0; r < 8; ++r)
      Slds[(r + half * 8) * S_PITCH + n0 + lm] = acc[r];
  }
  __syncthreads();

  // ---- phase 2: exact softmax stats per row (1024 wide) ----
  const int srow = tid & 15;
  const int chunk = tid >> 4;          // 8 chunks x 128 cols
  float* rowp = &Slds[srow * S_PITCH + chunk * 128];
  float mx = -INFINITY;
  for (int i = 0; i < 128; ++i) mx = fmaxf(mx, betaF * rowp[i]);
  red[srow * 8 + chunk] = mx;
  __syncthreads();
  if (tid < 16) {
    float m = red[tid * 8];
    for (int c = 1; c < 8; ++c) m = fmaxf(m, red[tid * 8 + c]);
    rmax[tid] = m;
  }
  __syncthreads();
  const float rmx = rmax[srow];
  float sum = 0.f;
  for (int i = 0; i < 128; ++i) {
    const float e = __expf(betaF * rowp[i] - rmx);
    rowp[i] = e;                       // S -> unnormalized P in place
    sum += e;
  }
  red[srow * 8 + chunk] = sum;
  __syncthreads();
  if (tid < 16) {
    float s = 0.f;
    for (int c = 0; c < 8; ++c) s += red[tid * 8 + c];
    rinv[tid] = 1.0f / s;              // fold normalization into output
  }
  __syncthreads();

  float ri[8];
  for (int r = 0; r < 8; ++r) ri[r] = rinv[r + half * 8];

  // ---- phase 3: O = P V ; wave wv owns output d-columns [wv*64, wv*64+64) ----
  for (int j = 0; j < 4; ++j) {
    const int d0 = (wv * 4 + j) * 16;
    v8f acc = {};
    for (int k = 0; k < N_TOK; k += 4) {
      // A = P[m][k..], from LDS (8B, bank-conflict-free via S_PITCH pad)
      v2f a = *(const v2f*)&Slds[lm * S_PITCH + k + koff];
      // B[kk][n] = V[kk][d0+n] = slice[(d0+n)*1024 + kk] -> contiguous pair
      v2f b = *(const v2f*)&slice[(d0 + lm) * N_TOK + k + koff];
      acc = __builtin_amdgcn_wmma_f32_16x16x4_f32(false, a, false, b,
                                                  (short)0, acc, false, false);
    }
    for (int r = 0; r < 8; ++r) {
      const int m = r + half * 8;
      outp[(long long)(m0 + m) * D_DIM + d0 + lm] = acc[r] * ri[r];
    }
  }
}

extern "C" void kernel_launch(void* const* d_in, const int* in_sizes, int n_in,
                              void* d_out, int out_size, void* d_ws, size_t ws_size,
                              hipStream_t stream) {
  const float* x   = (const float*)d_in[0];
  const int* beta  = (const int*)d_in[1];
  float* out       = (float*)d_out;
  const int BL = in_sizes[0] / (D_DIM * N_TOK);   // B*L = 20
  dim3 grid(N_TOK / 16, BL);                      // 64 row tiles x 20 slices
  dim3 block(128);
  const size_t shmem = SMEM_FLOATS * sizeof(float);  // ~83 KB of the 320 KB LDS
  attn_f32wmma_kernel<<<grid, block, shmem, stream>>>(x, beta, out);
}